// BioSelfAttention_45372034515554
// MI455X (gfx1250) — compile-verified
//
#include <hip/hip_runtime.h>

typedef __attribute__((ext_vector_type(2))) float v2f;
typedef __attribute__((ext_vector_type(8))) float v8f;

namespace {
constexpr int   kSteps   = 100;     // LIF_model_steps
constexpr float kLeak    = 0.05f;   // dt / tau_rc = 0.001 / 0.02
constexpr float kRateScl = 10.0f;   // 1 / (n_steps * dt)
constexpr int   kWta     = 20;      // WTA_STEPS
constexpr float kInh     = -0.9f;   // wta_inhibition
constexpr float kExcmInh = 2.0f;    // exc - inh = 1.1 - (-0.9)
constexpr int   kT       = 256;
constexpr int   kD       = 64;
}

// 100-step leaky integrate-and-fire; returns empirical firing rate.
// fp contract off so v + 0.05*(J-v) rounds mul-then-add like the reference
// (spike counts are threshold-discontinuous, so rounding order matters).
__device__ __forceinline__ float lif_rate(float J) {
  float v = 0.0f, cnt = 0.0f;
  {
#pragma clang fp contract(off)
    for (int s = 0; s < kSteps; ++s) {
      float v2 = v + kLeak * (J - v);
      if (v2 >= 1.0f) { cnt += 1.0f; v = 0.0f; } else { v = v2; }
    }
  }
  return cnt * kRateScl;
}

// ---------------------------------------------------------------------------
// K1: one block per (b,h) head, 256 threads (8 wave32).
//  - J[t] = <Q[t,:],K[t,:]> via V_WMMA_F32_16X16X4_F32 with B = ones
//    (rowsum-as-matmul; full f32 precision on the matrix pipe)
//  - per-token LIF, then 20x WTA over T=256 with LDS reductions.
// ---------------------------------------------------------------------------
__global__ __launch_bounds__(256) void k1_qk_lif_wta(
    const float* __restrict__ Q, const float* __restrict__ K,
    float* __restrict__ rates_out) {
  __shared__ float Jlds[kT];
  __shared__ float red[kT];

  const int head = blockIdx.x;
  const int tid  = threadIdx.x;
  const int lane = tid & 31;
  const int w    = tid >> 5;

  const float* __restrict__ Qh = Q + (size_t)head * kT * kD;
  const float* __restrict__ Kh = K + (size_t)head * kT * kD;

  // ISA 7.12.2, 32-bit A 16x4: lanes 0-15 hold (M=lane, K=0/1) in v0/v1,
  // lanes 16-31 hold (M=lane-16, K=2/3). B = all-ones is layout-invariant.
  const v2f ones = {1.0f, 1.0f};
  const int M  = lane & 15;
  const int kh = lane >> 4;           // 0: K=0,1   1: K=2,3

  for (int tile = 0; tile < 2; ++tile) {
    const int t0 = w * 32 + tile * 16;          // 16-token tile
    const int rowbase = (t0 + M) * kD + 2 * kh;
    v8f acc = {};
    for (int c = 0; c < 16; ++c) {              // 16 chunks of K=4 cover D=64
      const int off = rowbase + 4 * c;
      v2f a;
      a.x = Qh[off]     * Kh[off];
      a.y = Qh[off + 1] * Kh[off + 1];
      acc = __builtin_amdgcn_wmma_f32_16x16x4_f32(
          /*neg_a=*/false, a, /*neg_b=*/false, ones,
          /*c_mod=*/(short)0, acc, /*reuse_a=*/false, /*reuse_b=*/false);
    }
    // D layout: VGPR r, lanes 0-15 -> M=r ; lanes 16-31 -> M=8+r.
    // All 16 columns hold the same row-sum; lane 0 / lane 16 publish.
    if (lane == 0) {
      for (int r = 0; r < 8; ++r) Jlds[t0 + r] = acc[r];
    } else if (lane == 16) {
      for (int r = 0; r < 8; ++r) Jlds[t0 + 8 + r] = acc[r];
    }
  }
  __syncthreads();

  // per-token LIF -> firing rate
  float r = lif_rate(Jlds[tid]);

  // 20x WTA over the 256 tokens of this head
  for (int it = 0; it < kWta; ++it) {
    red[tid] = r; __syncthreads();
    for (int s = 128; s > 0; s >>= 1) {
      if (tid < s) red[tid] += red[tid + s];
      __syncthreads();
    }
    const float total = red[0]; __syncthreads();

    r = r + kExcmInh * r + kInh * total;

    red[tid] = r; __syncthreads();
    for (int s = 128; s > 0; s >>= 1) {
      if (tid < s) red[tid] = fmaxf(red[tid], red[tid + s]);
      __syncthreads();
    }
    const float mx = red[0]; __syncthreads();

    const float e = expf(r - mx);
    red[tid] = e; __syncthreads();
    for (int s = 128; s > 0; s >>= 1) {
      if (tid < s) red[tid] += red[tid + s];
      __syncthreads();
    }
    const float se = red[0]; __syncthreads();
    r = e / se;
  }
  rates_out[head * kT + tid] = r;
}

// ---------------------------------------------------------------------------
// K2: elementwise J_v = rate * V, 100-step LIF. 262144 independent chains
// spread across the whole chip; context rates written into d_out.
// ---------------------------------------------------------------------------
__global__ __launch_bounds__(256) void k2_v_lif(
    const float* __restrict__ V, const float* __restrict__ rates,
    float* __restrict__ ctx, int n) {
  const int i = blockIdx.x * 256 + threadIdx.x;
  if (i >= n) return;
  const float J = rates[i >> 6] * V[i];   // i>>6 == (b,h,t) since D=64
  ctx[i] = lif_rate(J);
}

// ---------------------------------------------------------------------------
// K3: one block per head, 1024 threads, 16 contiguous elements/thread;
// 20x WTA over the flattened T*D = 16384 units, in-place on d_out.
// ---------------------------------------------------------------------------
__global__ __launch_bounds__(1024) void k3_wta_flat(float* __restrict__ ctx) {
  __shared__ float red[1024];
  const int head = blockIdx.x;
  const int tid  = threadIdx.x;
  float* __restrict__ base = ctx + (size_t)head * (kT * kD) + tid * 16;

  float r[16];
#pragma unroll
  for (int i = 0; i < 16; ++i) r[i] = base[i];

  for (int it = 0; it < kWta; ++it) {
    float ls = 0.0f;
#pragma unroll
    for (int i = 0; i < 16; ++i) ls += r[i];
    red[tid] = ls; __syncthreads();
    for (int s = 512; s > 0; s >>= 1) {
      if (tid < s) red[tid] += red[tid + s];
      __syncthreads();
    }
    const float total = red[0]; __syncthreads();

#pragma unroll
    for (int i = 0; i < 16; ++i) r[i] = r[i] + kExcmInh * r[i] + kInh * total;

    float lm = -__builtin_inff();
#pragma unroll
    for (int i = 0; i < 16; ++i) lm = fmaxf(lm, r[i]);
    red[tid] = lm; __syncthreads();
    for (int s = 512; s > 0; s >>= 1) {
      if (tid < s) red[tid] = fmaxf(red[tid], red[tid + s]);
      __syncthreads();
    }
    const float mx = red[0]; __syncthreads();

    float e[16];
    float le = 0.0f;
#pragma unroll
    for (int i = 0; i < 16; ++i) { e[i] = expf(r[i] - mx); le += e[i]; }
    red[tid] = le; __syncthreads();
    for (int s = 512; s > 0; s >>= 1) {
      if (tid < s) red[tid] += red[tid + s];
      __syncthreads();
    }
    const float se = red[0]; __syncthreads();

#pragma unroll
    for (int i = 0; i < 16; ++i) r[i] = e[i] / se;
  }

#pragma unroll
  for (int i = 0; i < 16; ++i) base[i] = r[i];
}

extern "C" void kernel_launch(void* const* d_in, const int* in_sizes, int n_in,
                              void* d_out, int out_size, void* d_ws, size_t ws_size,
                              hipStream_t stream) {
  const float* Q = (const float*)d_in[0];
  const float* K = (const float*)d_in[1];
  const float* V = (const float*)d_in[2];
  float* out   = (float*)d_out;
  float* rates = (float*)d_ws;            // B*H*T floats (16 KB)

  const int n  = out_size;                // B*H*T*D = 262144
  const int BH = n / (kT * kD);           // 16 heads

  k1_qk_lif_wta<<<BH, 256, 0, stream>>>(Q, K, rates);
  k2_v_lif<<<(n + 255) / 256, 256, 0, stream>>>(V, rates, out, n);
  k3_wta_flat<<<BH, 1024, 0, stream>>>(out);
}